// HPPOActorCritic_17162689315477
// MI455X (gfx1250) — compile-verified
//
#include <hip/hip_runtime.h>
#include <cstddef>

typedef __attribute__((ext_vector_type(2))) float v2f;
typedef __attribute__((ext_vector_type(8))) float v8f;

#define KSPLIT 10   // 1000 / 10 = 100-wide K chunks (multiple of 4)

// Fixed 19-node graph; 36 directed edges + 19 self loops (SRC2/DST2).
__constant__ int c_SRC2[55] = {
  0,1,2,3,4,5,6,7,8,9,10,11,12,13,14,15,16,17,18,
  1,2,3,4,5,6,7,8,9,10,11,12,13,14,15,16,17,
  0,1,2,3,4,5,6,7,8,9,10,11,12,13,14,15,16,17,18};
__constant__ int c_DST2[55] = {
  1,2,3,4,5,6,7,8,9,10,11,12,13,14,15,16,17,18,17,
  0,1,2,3,4,5,6,7,8,9,10,11,12,13,14,15,16,
  0,1,2,3,4,5,6,7,8,9,10,11,12,13,14,15,16,17,18};

// ---------------------------- zero scratch ---------------------------------
__global__ void zero_buf(float* __restrict__ p, int n)
{
  int i = blockIdx.x * blockDim.x + threadIdx.x;
  if (i < n) p[i] = 0.f;
}

// ---------------- direct 5x5 stride-2 conv (NCHW, Cout=32) ----------------
__global__ void conv5x5_s2(const float* __restrict__ in, const float* __restrict__ w,
                           const float* __restrict__ b, float* __restrict__ out,
                           int Cin, int H, int W, int OH, int OW, int pad, int relu)
{
  int idx = blockIdx.x * blockDim.x + threadIdx.x;
  int total = 32 * OH * OW;
  if (idx >= total) return;
  int oc = idx / (OH * OW);
  int r  = idx % (OH * OW);
  int oy = r / OW, ox = r % OW;
  float s = b[oc];
  for (int ic = 0; ic < Cin; ++ic) {
    const float* wp = w + ((oc * Cin + ic) * 25);
    const float* ip = in + ic * H * W;
#pragma unroll
    for (int ky = 0; ky < 5; ++ky) {
      int iy = oy * 2 - pad + ky;
      if ((unsigned)iy >= (unsigned)H) continue;
#pragma unroll
      for (int kx = 0; kx < 5; ++kx) {
        int ix = ox * 2 - pad + kx;
        if ((unsigned)ix >= (unsigned)W) continue;
        s += ip[iy * W + ix] * wp[ky * 5 + kx];
      }
    }
  }
  if (relu) s = fmaxf(s, 0.0f);
  out[idx] = s;
}

// ------------- block-per-row matvec: out[o] = in . W[o,:] + b[o] -----------
// (bandwidth-bound path for fc0 150MB / fc1 2.4MB; float4 coalesced loads)
__global__ void matvec_row(const float* __restrict__ in, const float* __restrict__ W,
                           const float* __restrict__ bias, float* __restrict__ out, int K)
{
  int o = blockIdx.x, t = threadIdx.x;
  const float4* w4 = (const float4*)(W + (size_t)o * K);
  const float4* i4 = (const float4*)in;
  int K4 = K >> 2;   // K is a multiple of 4 for all callers
  float s = 0.f;
  for (int i = t; i < K4; i += 256) {
    float4 a = i4[i], b = w4[i];
    s += a.x * b.x + a.y * b.y + a.z * b.z + a.w * b.w;
  }
  __shared__ float red[256];
  red[t] = s; __syncthreads();
  for (int st = 128; st > 0; st >>= 1) {
    if (t < st) red[t] += red[t + st];
    __syncthreads();
  }
  if (t == 0) out[o] = red[0] + bias[o];
}

// ------------- thread-per-output matvec for small layers -------------------
__global__ void matvec_small(const float* __restrict__ in, const float* __restrict__ W,
                             const float* __restrict__ bias, float* __restrict__ out,
                             int M, int K)
{
  int o = blockIdx.x * blockDim.x + threadIdx.x;
  if (o >= M) return;
  const float* w = W + (size_t)o * K;
  float s = bias[o];
  for (int i = 0; i < K; ++i) s += in[i] * w[i];
  out[o] = s;
}

// ------------------------------- safe_norm ---------------------------------
__global__ void safenorm(const float* __restrict__ in, float* __restrict__ out, int n)
{
  __shared__ float smn[128], smx[128];
  int t = threadIdx.x;
  float mn = 3.0e38f, mx = -3.0e38f;
  for (int i = t; i < n; i += 128) { float v = in[i]; mn = fminf(mn, v); mx = fmaxf(mx, v); }
  smn[t] = mn; smx[t] = mx; __syncthreads();
  for (int s = 64; s > 0; s >>= 1) {
    if (t < s) { smn[t] = fminf(smn[t], smn[t + s]); smx[t] = fmaxf(smx[t], smx[t + s]); }
    __syncthreads();
  }
  float MN = smn[0], MX = smx[0];
  float d = MX - MN;
  bool small = fabsf(d) < 1e-6f;
  for (int i = t; i < n; i += 128) out[i] = small ? 0.f : (in[i] - MN) / d;
}

// -------- WMMA fp32 GEMM partials: part[kz] = A[:,kz-chunk] @ W^T ----------
// Fixed shape: [19,1000] @ [1000,1000]^T, KSPLIT K-chunks of 100.
// One wave per 16x16 tile per K-chunk (grid = 63 x 2 x KSPLIT). Invalid
// rows/cols redirect their base pointer to a zeroed scratch row, so the
// inner loop is pure load_b64 + v_wmma_f32_16x16x4_f32; compile-time trip
// count + unroll 5 lets the scheduler batch loads ahead of the WMMAs.
__global__ __launch_bounds__(32) void gemm_wmma_part(
    const float* __restrict__ A, const float* __restrict__ W,
    float* __restrict__ part, const float* __restrict__ zbuf)
{
  constexpr int M = 19, K = 1000, N = 1000, Kc = 1000 / KSPLIT;
  int lane = threadIdx.x;
  int n0 = blockIdx.x * 16;
  int mt = blockIdx.y;
  int kz = blockIdx.z;
  int sub = lane & 15;
  int hi  = lane >> 4;        // 0: K{0,1} / M rows 0-7 ; 1: K{2,3} / M rows 8-15
  int kh  = hi * 2;
  int m = mt * 16 + sub;
  int n = n0 + sub;
  const float* arow = ((m < M) ? (A + m * K) : zbuf) + kz * Kc;  // zbuf: >= K zeros
  const float* wrow = ((n < N) ? (W + n * K) : zbuf) + kz * Kc;
  v8f c = {0.f, 0.f, 0.f, 0.f, 0.f, 0.f, 0.f, 0.f};
#pragma unroll 5
  for (int k0 = 0; k0 < Kc; k0 += 4) {
    v2f a, b;
    a.x = arow[k0 + kh];        // A fragment: lanes 0-15 K{0,1}, 16-31 K{2,3}
    a.y = arow[k0 + kh + 1];
    b.x = wrow[k0 + kh];        // B fragment: B[k][n] = W[n][k]
    b.y = wrow[k0 + kh + 1];
    c = __builtin_amdgcn_wmma_f32_16x16x4_f32(
        /*neg_a=*/false, a, /*neg_b=*/false, b,
        /*c_mod=*/(short)0, c, /*reuse_a=*/false, /*reuse_b=*/false);
  }
  // C/D layout: VGPR r -> row (mt*16 + r + hi*8), col = n0 + (lane&15)
  int col = n0 + sub;
  int rbase = mt * 16 + hi * 8;
  float* pc = part + kz * (M * N);
  if (col < N) {
#pragma unroll
    for (int r2 = 0; r2 < 8; ++r2) {
      int row = rbase + r2;
      if (row < M) pc[row * N + col] = c[r2];
    }
  }
}

// ---- deterministic K-chunk reduction + epilogue: C (+)= sum(part)+b,relu --
__global__ void gemm_reduce(const float* __restrict__ part, float* __restrict__ C,
                            const float* __restrict__ bias, int acc, int relu)
{
  constexpr int M = 19, N = 1000;
  int idx = blockIdx.x * blockDim.x + threadIdx.x;
  if (idx >= M * N) return;
  float s = 0.f;
#pragma unroll
  for (int kz = 0; kz < KSPLIT; ++kz) s += part[kz * (M * N) + idx];
  if (acc) s += C[idx];
  if (bias) s += bias[idx % N];
  if (relu) s = fmaxf(s, 0.f);
  C[idx] = s;
}

// ---------------- SAGE layer 1 (F_in = 1), fused, with relu ----------------
__global__ void sage1(const float* __restrict__ xg, const float* __restrict__ wl,
                      const float* __restrict__ bl, const float* __restrict__ wr,
                      float* __restrict__ out)
{
  int idx = blockIdx.x * blockDim.x + threadIdx.x;
  if (idx >= 19 * 1000) return;
  int n = idx / 1000, o = idx % 1000;
  float agg = 0.f;
  for (int e = 0; e < 36; ++e)                 // raw edges only (no self loops)
    if (c_DST2[e] == n) agg += xg[c_SRC2[e]];
  float v = agg * wl[o] + bl[o] + xg[n] * wr[o];
  out[idx] = fmaxf(v, 0.f);
}

// ------------- neighbor sum over raw 36 edges: agg[n,f] --------------------
__global__ void sage_agg(const float* __restrict__ x, float* __restrict__ agg, int F)
{
  int idx = blockIdx.x * blockDim.x + threadIdx.x;
  if (idx >= 19 * F) return;
  int n = idx / F, f = idx % F;
  float s = 0.f;
  for (int e = 0; e < 36; ++e)
    if (c_DST2[e] == n) s += x[c_SRC2[e] * F + f];
  agg[idx] = s;
}

// ------------- GAT: per-node attention dot products ------------------------
__global__ void gat_dots(const float* __restrict__ xp, const float* __restrict__ a_s,
                         const float* __restrict__ a_d, float* __restrict__ asd,
                         float* __restrict__ add, int F)
{
  int n = blockIdx.x, t = threadIdx.x;
  float p = 0.f, q = 0.f;
  for (int f = t; f < F; f += 128) {
    float v = xp[n * F + f];
    p += v * a_s[f];
    q += v * a_d[f];
  }
  __shared__ float rp[128], rq[128];
  rp[t] = p; rq[t] = q; __syncthreads();
  for (int s = 64; s > 0; s >>= 1) {
    if (t < s) { rp[t] += rp[t + s]; rq[t] += rq[t + s]; }
    __syncthreads();
  }
  if (t == 0) { asd[n] = rp[0]; add[n] = rq[0]; }
}

// ------------- GAT: leaky-relu + segment softmax over DST2 -----------------
__global__ void gat_alpha(const float* __restrict__ asd, const float* __restrict__ add,
                          float* __restrict__ alpha)
{
  __shared__ float e[55];
  __shared__ float emax[19];
  __shared__ float esum[19];
  int t = threadIdx.x;
  if (t < 55) {
    float v = asd[c_SRC2[t]] + add[c_DST2[t]];
    e[t] = (v > 0.f) ? v : 0.2f * v;
  }
  __syncthreads();
  if (t < 19) {
    float m = -3.0e38f;
    for (int i = 0; i < 55; ++i) if (c_DST2[i] == t) m = fmaxf(m, e[i]);
    emax[t] = m;
  }
  __syncthreads();
  if (t < 55) e[t] = expf(e[t] - emax[c_DST2[t]]);
  __syncthreads();
  if (t < 19) {
    float s = 0.f;
    for (int i = 0; i < 55; ++i) if (c_DST2[i] == t) s += e[i];
    esum[t] = s;
  }
  __syncthreads();
  if (t < 55) alpha[t] = e[t] / esum[c_DST2[t]];
}

// ------------- GCN improved=True symmetric normalization -------------------
__global__ void gcn_norm(float* __restrict__ norm)
{
  __shared__ float dis[19];
  int t = threadIdx.x;
  if (t < 19) {
    float d = 0.f;
    for (int e = 0; e < 55; ++e)
      if (c_DST2[e] == t) d += (e < 36) ? 1.f : 2.f;   // self-loop weight = 2
    dis[t] = (d > 0.f) ? (1.0f / sqrtf(d)) : 0.f;
  }
  __syncthreads();
  if (t < 55) norm[t] = dis[c_SRC2[t]] * ((t < 36) ? 1.f : 2.f) * dis[c_DST2[t]];
}

// ------ weighted segment-sum over SRC2/DST2: out[n,f] = sum coef*xp + b ----
__global__ void edge_agg(const float* __restrict__ xp, const float* __restrict__ coef,
                         const float* __restrict__ bias, float* __restrict__ out,
                         int F, int relu)
{
  int idx = blockIdx.x * blockDim.x + threadIdx.x;
  if (idx >= 19 * F) return;
  int n = idx / F, f = idx % F;
  float s = bias[f];
  for (int e = 0; e < 55; ++e)
    if (c_DST2[e] == n) s += coef[e] * xp[c_SRC2[e] * F + f];
  if (relu) s = fmaxf(s, 0.f);
  out[idx] = s;
}

// ------------------------- mean over the 19 nodes --------------------------
__global__ void mean_nodes(const float* __restrict__ g, float* __restrict__ out, int F)
{
  int f = blockIdx.x * blockDim.x + threadIdx.x;
  if (f >= F) return;
  float s = 0.f;
  for (int n = 0; n < 19; ++n) s += g[n * F + f];
  out[f] = s * (1.0f / 19.0f);
}

extern "C" void kernel_launch(void* const* d_in, const int* in_sizes, int n_in,
                              void* d_out, int out_size, void* d_ws, size_t ws_size,
                              hipStream_t stream)
{
  (void)in_sizes; (void)n_in; (void)out_size; (void)ws_size;
  const float* x     = (const float*)d_in[0];
  const float* state = (const float*)d_in[1];
  const float* xg    = (const float*)d_in[2];
  const float* c1w = (const float*)d_in[3];  const float* c1b = (const float*)d_in[4];
  const float* c2w = (const float*)d_in[5];  const float* c2b = (const float*)d_in[6];
  const float* c3w = (const float*)d_in[7];  const float* c3b = (const float*)d_in[8];
  const float* fc0w = (const float*)d_in[9];  const float* fc0b = (const float*)d_in[10];
  const float* fc1w = (const float*)d_in[11]; const float* fc1b = (const float*)d_in[12];
  const float* fc2w = (const float*)d_in[13]; const float* fc2b = (const float*)d_in[14];
  const float* fc3w = (const float*)d_in[15]; const float* fc3b = (const float*)d_in[16];
  const float* s1l1_wl = (const float*)d_in[17]; const float* s1l1_bl = (const float*)d_in[18];
  const float* s1l1_wr = (const float*)d_in[19];
  const float* s1l2_wl = (const float*)d_in[20]; const float* s1l2_bl = (const float*)d_in[21];
  const float* s1l2_wr = (const float*)d_in[22];
  const float* g2w = (const float*)d_in[23]; const float* g2as = (const float*)d_in[24];
  const float* g2ad = (const float*)d_in[25]; const float* g2b = (const float*)d_in[26];
  const float* s3l1_wl = (const float*)d_in[27]; const float* s3l1_bl = (const float*)d_in[28];
  const float* s3l1_wr = (const float*)d_in[29];
  const float* s3l2_wl = (const float*)d_in[30]; const float* s3l2_bl = (const float*)d_in[31];
  const float* s3l2_wr = (const float*)d_in[32];
  const float* g4w = (const float*)d_in[33]; const float* g4as = (const float*)d_in[34];
  const float* g4ad = (const float*)d_in[35]; const float* g4b = (const float*)d_in[36];
  const float* g5w = (const float*)d_in[37]; const float* g5b = (const float*)d_in[38];
  const float* fcgw = (const float*)d_in[39]; const float* fcgb = (const float*)d_in[40];
  const float* fc4w = (const float*)d_in[41]; const float* fc4b = (const float*)d_in[42];
  const float* actw = (const float*)d_in[43]; const float* actb = (const float*)d_in[44];
  const float* criw = (const float*)d_in[45]; const float* crib = (const float*)d_in[46];
  float* out = (float*)d_out;

  // workspace carve-up (~1.6 MB of fp32)
  float* ws = (float*)d_ws;
  float* h1    = ws; ws += 32 * 61 * 61;
  float* h2    = ws; ws += 32 * 29 * 29;
  float* h3    = ws; ws += 6272;
  float* v0    = ws; ws += 6000;
  float* v1    = ws; ws += 100;
  float* s0    = ws; ws += 100;
  float* s1b   = ws; ws += 100;
  float* fcat  = ws; ws += 300;
  float* f4    = ws; ws += 200;
  float* gA    = ws; ws += 19 * 1000;
  float* gB    = ws; ws += 19 * 1000;
  float* gT    = ws; ws += 19 * 1000;
  float* asd   = ws; ws += 19;
  float* add   = ws; ws += 19;
  float* coef  = ws; ws += 55;
  float* gmean = ws; ws += 1000;
  float* zbuf  = ws; ws += 1000;                 // zero row for OOB lanes
  float* part  = ws; ws += KSPLIT * 19 * 1000;   // per-K-chunk GEMM partials

  zero_buf<<<4, 256, 0, stream>>>(zbuf, 1000);

  // ---------------- CNN branch ----------------
  conv5x5_s2<<<(32 * 61 * 61 + 255) / 256, 256, 0, stream>>>(x,  c1w, c1b, h1, 1,  123, 123, 61, 61, 1, 1);
  conv5x5_s2<<<(32 * 29 * 29 + 255) / 256, 256, 0, stream>>>(h1, c2w, c2b, h2, 32, 61,  61,  29, 29, 0, 1);
  conv5x5_s2<<<(32 * 14 * 14 + 255) / 256, 256, 0, stream>>>(h2, c3w, c3b, h3, 32, 29,  29,  14, 14, 1, 0);
  matvec_row<<<6000, 256, 0, stream>>>(h3, fc0w, fc0b, v0, 6272);   // 150 MB, BW-bound
  matvec_row<<<100,  256, 0, stream>>>(v0, fc1w, fc1b, v1, 6000);
  safenorm<<<1, 128, 0, stream>>>(v1, fcat + 0, 100);

  // ---------------- state branch ----------------
  matvec_small<<<1, 128, 0, stream>>>(state, fc2w, fc2b, s0, 100, 20);
  matvec_small<<<1, 128, 0, stream>>>(s0,    fc3w, fc3b, s1b, 100, 100);
  safenorm<<<1, 128, 0, stream>>>(s1b, fcat + 100, 100);

  // ---------------- graph branch ----------------
  const int GN = (19 * 1000 + 255) / 256;
  dim3 gg(63, 2, KSPLIT), gb(32);   // 63 N-tiles x 2 M-tiles x 10 K-chunks
  // GEMM helper macro: dst (+)= src @ W^T (+ bias, relu), via partials
#define WGEMM(SRC, WMAT, DST, BIAS, ACC, RELU)                                 \
  do {                                                                         \
    gemm_wmma_part<<<gg, gb, 0, stream>>>(SRC, WMAT, part, zbuf);              \
    gemm_reduce<<<GN, 256, 0, stream>>>(part, DST, BIAS, ACC, RELU);           \
  } while (0)

  // SAGE s1l1 (1 -> 1000) + relu
  sage1<<<GN, 256, 0, stream>>>(xg, s1l1_wl, s1l1_bl, s1l1_wr, gA);
  // SAGE s1l2: gB = agg@wl^T + gA@wr^T + bl, relu
  sage_agg<<<GN, 256, 0, stream>>>(gA, gT, 1000);
  WGEMM(gT, s1l2_wl, gB, nullptr, 0, 0);
  WGEMM(gA, s1l2_wr, gB, s1l2_bl, 1, 1);
  // GAT g2
  WGEMM(gB, g2w, gT, nullptr, 0, 0);
  gat_dots<<<19, 128, 0, stream>>>(gT, g2as, g2ad, asd, add, 1000);
  gat_alpha<<<1, 64, 0, stream>>>(asd, add, coef);
  edge_agg<<<GN, 256, 0, stream>>>(gT, coef, g2b, gA, 1000, 1);
  // SAGE s3l1
  sage_agg<<<GN, 256, 0, stream>>>(gA, gT, 1000);
  WGEMM(gT, s3l1_wl, gB, nullptr, 0, 0);
  WGEMM(gA, s3l1_wr, gB, s3l1_bl, 1, 1);
  // SAGE s3l2
  sage_agg<<<GN, 256, 0, stream>>>(gB, gT, 1000);
  WGEMM(gT, s3l2_wl, gA, nullptr, 0, 0);
  WGEMM(gB, s3l2_wr, gA, s3l2_bl, 1, 1);
  // GAT g4
  WGEMM(gA, g4w, gT, nullptr, 0, 0);
  gat_dots<<<19, 128, 0, stream>>>(gT, g4as, g4ad, asd, add, 1000);
  gat_alpha<<<1, 64, 0, stream>>>(asd, add, coef);
  edge_agg<<<GN, 256, 0, stream>>>(gT, coef, g4b, gB, 1000, 1);
  // GCN g5 (no relu)
  gcn_norm<<<1, 64, 0, stream>>>(coef);
  WGEMM(gB, g5w, gT, nullptr, 0, 0);
  edge_agg<<<GN, 256, 0, stream>>>(gT, coef, g5b, gA, 1000, 0);
  mean_nodes<<<4, 256, 0, stream>>>(gA, gmean, 1000);
  matvec_small<<<1, 128, 0, stream>>>(gmean, fcgw, fcgb, v1, 100, 1000);
  safenorm<<<1, 128, 0, stream>>>(v1, fcat + 200, 100);

  // ---------------- heads ----------------
  matvec_small<<<1, 256, 0, stream>>>(fcat, fc4w, fc4b, f4, 200, 300);
  matvec_small<<<1, 32, 0, stream>>>(f4, actw, actb, out, 19, 200);       // logits[19]
  matvec_small<<<1, 32, 0, stream>>>(f4, criw, crib, out + 19, 1, 200);   // value[1]
#undef WGEMM
}